// Encoder_87179246174334
// MI455X (gfx1250) — compile-verified
//
#include <hip/hip_runtime.h>

// Problem constants (from the reference)
#define N0C 200000
#define N1C 50000
#define N2C 10000
#define E0C 1600000
#define E1C 320000

typedef __attribute__((ext_vector_type(2))) float v2f;
typedef __attribute__((ext_vector_type(8))) float v8f;

// ---------------------------------------------------------------------------
// One wave32 per edge: gather a 256-f32 feature row and atomically accumulate
// it into agg[dst], plus a degree counter. 8 floats per lane (2x float4 load).
// ---------------------------------------------------------------------------
__global__ void __launch_bounds__(256) scatter_mean_kernel(
    const float* __restrict__ feat, const int* __restrict__ src,
    const int* __restrict__ dst, float* __restrict__ agg,
    float* __restrict__ cnt, int E)
{
    const int gw   = (blockIdx.x * 256 + threadIdx.x) >> 5;  // global wave id
    const int lane = threadIdx.x & 31;
    if (gw >= E) return;  // uniform per wave

    const int s = src[gw];
    const int d = dst[gw];

    const float4* fr = reinterpret_cast<const float4*>(feat + (size_t)s * 256);
    const float4 a = fr[lane * 2 + 0];
    const float4 b = fr[lane * 2 + 1];

    float* ag = agg + (size_t)d * 256 + lane * 8;
    atomicAdd(ag + 0, a.x); atomicAdd(ag + 1, a.y);
    atomicAdd(ag + 2, a.z); atomicAdd(ag + 3, a.w);
    atomicAdd(ag + 4, b.x); atomicAdd(ag + 5, b.y);
    atomicAdd(ag + 6, b.z); atomicAdd(ag + 7, b.w);

    if (lane == 0) atomicAdd(cnt + d, 1.0f);
}

// ---------------------------------------------------------------------------
// Fused SAGE layer:
//   out = relu( mean@Wl + bl + xdst@Wr + 0.25*(attr[batch]@Wa + ba) )
// Implemented as one K=768 GEMM: A = [mean | xdst | 0.25*attr_g] staged in
// LDS; B = the three 256xN weight blocks read from L2. One 16-row M tile per
// block; each wave owns one 16x16 output tile and runs 192
// v_wmma_f32_16x16x4_f32 accumulations (full fp32 precision).
// ---------------------------------------------------------------------------
template <int NCOLS>
__global__ void __launch_bounds__(NCOLS * 2) sage_layer_kernel(
    const float* __restrict__ agg, const float* __restrict__ cnt,
    const float* __restrict__ xdst, const float* __restrict__ attr,
    const int* __restrict__ batch,
    const float* __restrict__ Wl, const float* __restrict__ bl,
    const float* __restrict__ Wr,
    const float* __restrict__ Wa, const float* __restrict__ ba,
    float* __restrict__ out)
{
    constexpr int LDA = 768 + 4;  // +4 pad: row stride 772 % 64 banks = 4
    __shared__ float As[16][LDA];

    const int m0 = blockIdx.x * 16;

    // Stage A tile: [mean | xdst | 0.25*attr[batch]]  (16 x 768 f32)
    for (int idx = threadIdx.x; idx < 16 * 256; idx += blockDim.x) {
        const int r = idx >> 8;
        const int c = idx & 255;
        const int gm = m0 + r;
        const float invc = 1.0f / fmaxf(cnt[gm], 1.0f);
        As[r][c]       = agg[(size_t)gm * 256 + c] * invc;
        As[r][256 + c] = xdst[(size_t)gm * 256 + c];
        As[r][512 + c] = 0.25f * attr[(size_t)batch[gm] * 256 + c];
    }
    __syncthreads();

    const int wave = threadIdx.x >> 5;
    const int lane = threadIdx.x & 31;
    const int col  = wave * 16 + (lane & 15);
    const int koff = (lane & 16) ? 2 : 0;   // A/B VGPR K split across lane halves
    const int arow = lane & 15;

    v8f acc = {};
    const float* Wseg[3] = { Wl, Wr, Wa };

    #pragma unroll
    for (int seg = 0; seg < 3; ++seg) {
        const float* __restrict__ W  = Wseg[seg];
        const float* __restrict__ Ap = &As[arow][seg * 256];
        #pragma unroll 8
        for (int k = 0; k < 256; k += 4) {
            // A fragment: two consecutive K values (8B-aligned LDS load)
            const float2 av = *reinterpret_cast<const float2*>(Ap + k + koff);
            v2f a; a.x = av.x; a.y = av.y;
            // B fragment: same K rows of the weight block, column = col
            v2f b;
            b.x = W[(size_t)(k + koff)     * NCOLS + col];
            b.y = W[(size_t)(k + koff + 1) * NCOLS + col];
            acc = __builtin_amdgcn_wmma_f32_16x16x4_f32(
                false, a, false, b, (short)0, acc, false, false);
        }
    }

    // C/D layout: lanes 0-15 -> M = vgpr, lanes 16-31 -> M = vgpr + 8
    const float bias = bl[col] + 0.25f * ba[col];
    const int mhi = (lane & 16) ? 8 : 0;
    #pragma unroll
    for (int v = 0; v < 8; ++v) {
        const float val = acc[v] + bias;
        out[(size_t)(m0 + mhi + v) * NCOLS + col] = fmaxf(val, 0.0f);
    }
}

// ---------------------------------------------------------------------------
extern "C" void kernel_launch(void* const* d_in, const int* in_sizes, int n_in,
                              void* d_out, int out_size, void* d_ws, size_t ws_size,
                              hipStream_t stream)
{
    (void)in_sizes; (void)n_in; (void)out_size; (void)ws_size;

    const float* x      = (const float*)d_in[0];
    const float* attr   = (const float*)d_in[1];
    const int*   src0   = (const int*)d_in[2];
    const int*   dst0   = (const int*)d_in[3];
    const int*   src1   = (const int*)d_in[4];
    const int*   dst1   = (const int*)d_in[5];
    const int*   batch0 = (const int*)d_in[6];
    const int*   batch1 = (const int*)d_in[7];
    const float* Wl0 = (const float*)d_in[8];
    const float* bl0 = (const float*)d_in[9];
    const float* Wr0 = (const float*)d_in[10];
    const float* Wl1 = (const float*)d_in[11];
    const float* bl1 = (const float*)d_in[12];
    const float* Wr1 = (const float*)d_in[13];
    const float* Wa0 = (const float*)d_in[14];
    const float* ba0 = (const float*)d_in[15];
    const float* Wa1 = (const float*)d_in[16];
    const float* ba1 = (const float*)d_in[17];

    // Workspace layout (f32): agg0 | cnt0 | h | agg1 | cnt1  (~113 MB)
    float* ws   = (float*)d_ws;
    float* agg0 = ws;
    float* cnt0 = agg0 + (size_t)N1C * 256;
    float* h    = cnt0 + N1C;
    float* agg1 = h + (size_t)N1C * 256;
    float* cnt1 = agg1 + (size_t)N2C * 256;

    hipMemsetAsync(agg0, 0, (size_t)N1C * 256 * sizeof(float), stream);
    hipMemsetAsync(cnt0, 0, (size_t)N1C * sizeof(float), stream);
    hipMemsetAsync(agg1, 0, (size_t)N2C * 256 * sizeof(float), stream);
    hipMemsetAsync(cnt1, 0, (size_t)N2C * sizeof(float), stream);

    // Layer 0
    scatter_mean_kernel<<<E0C / 8, 256, 0, stream>>>(x, src0, dst0, agg0, cnt0, E0C);
    sage_layer_kernel<256><<<N1C / 16, 512, 0, stream>>>(
        agg0, cnt0, x, attr, batch0, Wl0, bl0, Wr0, Wa0, ba0, h);

    // Layer 1
    scatter_mean_kernel<<<E1C / 8, 256, 0, stream>>>(h, src1, dst1, agg1, cnt1, E1C);
    sage_layer_kernel<128><<<N2C / 16, 256, 0, stream>>>(
        agg1, cnt1, h, attr, batch1, Wl1, bl1, Wr1, Wa1, ba1, (float*)d_out);
}